// sub_GMN_45758581571766
// MI455X (gfx1250) — compile-verified
//
#include <hip/hip_runtime.h>
#include <math.h>
#include <stdint.h>

// Problem sizes (fixed by the reference)
#define B_    64
#define NQ_   32
#define ND_   512
#define DIN_  64
#define D_    128
#define K_    16

typedef float v2f __attribute__((ext_vector_type(2)));
typedef float v8f __attribute__((ext_vector_type(8)));

// ---------------------------------------------------------------------------
// CDNA5 async global->LDS copy (ASYNCcnt-tracked), ISA §10.7 / §15.18.3.
// dst VGPR holds the per-lane LDS byte address; 64-bit global address in VGPRs.
// ---------------------------------------------------------------------------
__device__ __forceinline__ void async_copy_b128(uint32_t lds_off, const float* g) {
  asm volatile("global_load_async_to_lds_b128 %0, %1, off"
               :: "v"(lds_off), "v"(g) : "memory");
}
__device__ __forceinline__ void wait_asynccnt0() {
  asm volatile("s_wait_asynccnt 0x0" ::: "memory");
}

// ---------------------------------------------------------------------------
// WMMA helpers: V_WMMA_F32_16X16X4_F32, wave32, one 16x16 f32 tile per wave.
// A tile (16x4): lane l<16 -> M=l, K={0,1}; lane>=16 -> M=l-16, K={2,3}
// C/D tile: vgpr v -> row (l>>4)*8+v, col l&15
// All strides are compile-time so the K-loop folds into load immediates.
// ---------------------------------------------------------------------------
template <int LDA, int LDB, int KTOT>
__device__ __forceinline__ v8f wmma_tile_nn(const float* A, const float* Bm,
                                            v8f acc) {
  const int l  = threadIdx.x & 31;
  const int mn = l & 15;
  const int kh = (l >> 4) << 1;
  const float* Ar = A + mn * LDA + kh;   // one base, constant offsets below
  const float* Bc = Bm + kh * LDB + mn;
#pragma unroll
  for (int k0 = 0; k0 < KTOT; k0 += 4) {
    v2f a, b;
    a.x = Ar[k0 + 0];
    a.y = Ar[k0 + 1];
    b.x = Bc[k0 * LDB + 0 * LDB];
    b.y = Bc[k0 * LDB + 1 * LDB];
    acc = __builtin_amdgcn_wmma_f32_16x16x4_f32(false, a, false, b, (short)0,
                                                acc, false, false);
  }
  return acc;
}

// C = A * Bt^T, Bt is [16 x KTOT] row-major (rows are the N dimension)
template <int LDA, int LDBT, int KTOT>
__device__ __forceinline__ v8f wmma_tile_nt(const float* A, const float* Bt,
                                            v8f acc) {
  const int l  = threadIdx.x & 31;
  const int mn = l & 15;
  const int kh = (l >> 4) << 1;
  const float* Ar = A + mn * LDA + kh;
  const float* Br = Bt + mn * LDBT + kh;
#pragma unroll
  for (int k0 = 0; k0 < KTOT; k0 += 4) {
    v2f a, b;
    a.x = Ar[k0 + 0];
    a.y = Ar[k0 + 1];
    b.x = Br[k0 + 0];
    b.y = Br[k0 + 1];
    acc = __builtin_amdgcn_wmma_f32_16x16x4_f32(false, a, false, b, (short)0,
                                                acc, false, false);
  }
  return acc;
}

// Same NT tile product but with the B (16 x KTOT row-major) tile in LDS.
template <int LDA, int KTOT>
__device__ __forceinline__ v8f wmma_tile_nt_lds(const float* A,
                                                const float* __restrict__ sBt,
                                                v8f acc) {
  const int l  = threadIdx.x & 31;
  const int mn = l & 15;
  const int kh = (l >> 4) << 1;
  const float* Ar = A + mn * LDA + kh;
  const float* Br = sBt + mn * KTOT + kh;
#pragma unroll
  for (int k0 = 0; k0 < KTOT; k0 += 4) {
    v2f a, b;
    a.x = Ar[k0 + 0];
    a.y = Ar[k0 + 1];
    b.x = Br[k0 + 0];                    // ds_load_b64, constant offset
    b.y = Br[k0 + 1];
    acc = __builtin_amdgcn_wmma_f32_16x16x4_f32(false, a, false, b, (short)0,
                                                acc, false, false);
  }
  return acc;
}

template <int LDC>
__device__ __forceinline__ void wmma_store_tile(float* C, v8f d) {
  const int l  = threadIdx.x & 31;
  const int n  = l & 15;
  const int mb = (l >> 4) << 3;
  float* Cr = C + (size_t)mb * LDC + n;
#pragma unroll
  for (int v = 0; v < 8; ++v) Cr[v * LDC] = d[v];
}

// ---------------------------------------------------------------------------
// GCN kernels
// ---------------------------------------------------------------------------
__global__ void k_fill(float* p, float v, int n) {
  int i = blockIdx.x * blockDim.x + threadIdx.x;
  if (i < n) p[i] = v;
}

__global__ void k_deg(const int* dst, float* deg, int E) {
  int i = blockIdx.x * blockDim.x + threadIdx.x;
  if (i < E) atomicAdd(&deg[dst[i]], 1.0f);
}

__global__ void k_dinv(float* deg, int n) {  // in place deg -> deg^-0.5
  int i = blockIdx.x * blockDim.x + threadIdx.x;
  if (i < n) deg[i] = rsqrtf(deg[i]);
}

// C[M,D_] = A[M,KTOT] * B[KTOT,D_], one wave per 16x16 tile
template <int KTOT>
__global__ void k_gemm_nn(const float* A, const float* Bm, float* C) {
  const float* At = A + (size_t)blockIdx.x * 16 * KTOT;
  const float* Bt = Bm + (size_t)blockIdx.y * 16;
  float* Ct = C + (size_t)blockIdx.x * 16 * D_ + (size_t)blockIdx.y * 16;
  v8f acc = {};
  acc = wmma_tile_nn<KTOT, D_, KTOT>(At, Bt, acc);
  wmma_store_tile<D_>(Ct, acc);
}

// Out[n,c] = dinv[n]^2 * H[n,c] + bias[c]
__global__ void k_gcn_self(const float* H, const float* dinv, const float* bias,
                           float* Out, int tot) {
  int i = blockIdx.x * blockDim.x + threadIdx.x;
  if (i >= tot) return;
  int n = i >> 7, c = i & 127;
  float dv = dinv[n];
  Out[i] = dv * dv * H[i] + bias[c];
}

// Out[t,c] += dinv[s]*dinv[t]*H[s,c] for every edge
__global__ void k_gcn_edge(const int* src, const int* dst, const float* H,
                           const float* dinv, float* Out, int E) {
  int i = blockIdx.x * blockDim.x + threadIdx.x;
  if (i >= E * D_) return;
  int e = i >> 7, c = i & 127;
  int s = src[e], t = dst[e];
  atomicAdd(&Out[(size_t)t * D_ + c], dinv[s] * dinv[t] * H[(size_t)s * D_ + c]);
}

__global__ void k_relu(float* p, int n) {
  int i = blockIdx.x * blockDim.x + threadIdx.x;
  if (i < n) p[i] = fmaxf(p[i], 0.0f);
}

// ---------------------------------------------------------------------------
// Level kernels
// ---------------------------------------------------------------------------
// att logits: L[b,q,n] = (q . d) / sqrt(D); grid(NQ/16, ND/16, B)
__global__ void k_att_logits(const float* Q, const float* Dd, float* L) {
  int qt = blockIdx.x, nt = blockIdx.y, b = blockIdx.z;
  const float* A  = Q  + ((size_t)b * NQ_ + qt * 16) * D_;
  const float* Bt = Dd + ((size_t)b * ND_ + nt * 16) * D_;
  v8f acc = {};
  acc = wmma_tile_nt<D_, D_, D_>(A, Bt, acc);
  const float s = 0.08838834764831845f;  // 1/sqrt(128)
#pragma unroll
  for (int v = 0; v < 8; ++v) acc[v] *= s;
  wmma_store_tile<ND_>(L + ((size_t)b * NQ_ + qt * 16) * ND_ + nt * 16, acc);
}

// in-place row softmax over 512 columns; grid = rows, block = 256
__global__ void k_softmax_rows512(float* p) {
  __shared__ float red[256];
  float* row = p + (size_t)blockIdx.x * 512;
  int t = threadIdx.x;
  float a = row[t], b = row[t + 256];
  red[t] = fmaxf(a, b);
  __syncthreads();
  for (int s = 128; s > 0; s >>= 1) {
    if (t < s) red[t] = fmaxf(red[t], red[t + s]);
    __syncthreads();
  }
  float m = red[0];
  __syncthreads();
  float ea = expf(a - m), eb = expf(b - m);
  red[t] = ea + eb;
  __syncthreads();
  for (int s = 128; s > 0; s >>= 1) {
    if (t < s) red[t] += red[t + s];
    __syncthreads();
  }
  float inv = 1.0f / red[0];
  row[t] = ea * inv;
  row[t + 256] = eb * inv;
}

// lq[b,k,q] = sum_d Vn[k,d]*Q[b,q,d]
__global__ void k_lq(const float* Vn, const float* Q, float* lq) {
  int i = blockIdx.x * blockDim.x + threadIdx.x;
  if (i >= B_ * K_ * NQ_) return;
  int q = i % NQ_, k = (i / NQ_) % K_, b = i / (NQ_ * K_);
  const float* v = Vn + (size_t)k * 2 * D_;
  const float* x = Q + ((size_t)b * NQ_ + q) * D_;
  float s = 0.0f;
  for (int d = 0; d < D_; ++d) s += v[d] * x[d];
  lq[i] = s;
}

// ld[b,k,n] = sum_d Vn[k,D+d]*Dd[b,n,d]
__global__ void k_ld(const float* Vn, const float* Dd, float* ld) {
  int i = blockIdx.x * blockDim.x + threadIdx.x;
  if (i >= B_ * K_ * ND_) return;
  int n = i % ND_, k = (i / ND_) % K_, b = i / (ND_ * K_);
  const float* v = Vn + (size_t)k * 2 * D_ + D_;
  const float* x = Dd + ((size_t)b * ND_ + n) * D_;
  float s = 0.0f;
  for (int d = 0; d < D_; ++d) s += v[d] * x[d];
  ld[i] = s;
}

// qW[b,k,q,e] = sum_d Q[b,q,d]*Wn[k,d,e]; grid(NQ/16, D/16, B*K)
__global__ void k_qw(const float* Q, const float* Wn, float* qW) {
  int qt = blockIdx.x, et = blockIdx.y;
  int b = blockIdx.z / K_, k = blockIdx.z % K_;
  const float* A  = Q + ((size_t)b * NQ_ + qt * 16) * D_;
  const float* Bm = Wn + (size_t)k * D_ * D_ + et * 16;
  float* C = qW + (((size_t)b * K_ + k) * NQ_ + qt * 16) * D_ + et * 16;
  v8f acc = {};
  acc = wmma_tile_nn<D_, D_, D_>(A, Bm, acc);
  wmma_store_tile<D_>(C, acc);
}

// Fused NTN level: the shared d-tile (16x128 f32, 8KB) is staged into LDS via
// CDNA5 async global->LDS B128 copies (one per thread), ASYNCcnt-waited, then
// all 16 k-waves feed their WMMA bil tiles from LDS. he is formed in-register
// and reduced over k with LDS float atomics into eraw (cw-weighted, +cb) and
// dir (w_end-weighted he contribution). grid(NQ/16, ND/16, B), block 512.
__global__ void k_level(const float* qW, const float* Dd, const float* att,
                        const float* lq, const float* ld, const float* bn,
                        const float* cw, const float* cb, const float* w_end,
                        int level, float* eraw, float* dir) {
  __shared__ float sB[16 * D_];     // d tile [n within tile][e]
  __shared__ float s_eraw[256];
  __shared__ float s_dir[256];
  int qt = blockIdx.x, nt = blockIdx.y, b = blockIdx.z;
  int tid = threadIdx.x;
  int k = tid >> 5;

  // async stage of the d tile: 512 threads x 16B
  {
    int r  = tid >> 5;            // 0..15 : row (n within tile)
    int c4 = (tid & 31) * 4;      // 0..124 : col (e), 4 floats per thread
    const float* g = Dd + ((size_t)b * ND_ + nt * 16 + r) * D_ + c4;
    uint32_t loff = (uint32_t)(uintptr_t)(const void*)&sB[r * D_ + c4];
    async_copy_b128(loff, g);
    wait_asynccnt0();
  }
  if (tid < 256) { s_eraw[tid] = 0.0f; s_dir[tid] = 0.0f; }
  __syncthreads();

  const float* A = qW + (((size_t)b * K_ + k) * NQ_ + qt * 16) * D_;
  v8f acc = {};
  acc = wmma_tile_nt_lds<D_, D_>(A, sB, acc);

  const int l  = tid & 31;
  const int n  = l & 15;
  const int mb = (l >> 4) << 3;
  float cwk = cw[k];
  float bnk = bn[k];
  float wdir = 0.0f;
  if (level == 2)      wdir = w_end[19 + k];
  else if (level == 3) wdir = w_end[3 + k] + w_end[35 + k];

#pragma unroll
  for (int v = 0; v < 8; ++v) {
    int row = mb + v;                      // q within tile
    int q = qt * 16 + row;
    int nn = nt * 16 + n;
    float lqv = lq[((size_t)b * K_ + k) * NQ_ + q];
    float ldv = ld[((size_t)b * K_ + k) * ND_ + nn];
    float av  = att[((size_t)b * NQ_ + q) * ND_ + nn];
    float x = acc[v] + lqv + ldv + bnk;
    float he = (1.0f / (1.0f + expf(-x))) * av;
    int idx = row * 16 + n;
    atomicAdd(&s_eraw[idx], cwk * he);     // ds_add_f32
    atomicAdd(&s_dir[idx], wdir * he);
  }
  __syncthreads();
  if (tid < 256) {
    int row = tid >> 4, col = tid & 15;
    size_t o = ((size_t)b * NQ_ + qt * 16 + row) * ND_ + nt * 16 + col;
    eraw[o] = s_eraw[tid] + cb[0];
    if (dir) dir[o] = s_dir[tid];
  }
}

// out = w0*sm(e1) + w1*e2raw + w2*sm(e3) + dir2 + dir3 + b_end (pre-softmax)
__global__ void k_combine(const float* e1s, const float* e2, const float* e3s,
                          const float* dir2, const float* dir3,
                          const float* w_end, const float* b_end,
                          float* out, int tot) {
  int i = blockIdx.x * blockDim.x + threadIdx.x;
  if (i >= tot) return;
  out[i] = w_end[0] * e1s[i] + w_end[1] * e2[i] + w_end[2] * e3s[i] +
           dir2[i] + dir3[i] + b_end[0];
}

// ---------------------------------------------------------------------------
extern "C" void kernel_launch(void* const* d_in, const int* in_sizes, int n_in,
                              void* d_out, int out_size, void* d_ws, size_t ws_size,
                              hipStream_t stream) {
  const float* x_d = (const float*)d_in[0];
  const float* x_q = (const float*)d_in[1];
  const int* ei_d = (const int*)d_in[2];
  const int* ei_q = (const int*)d_in[3];
  const int Ed = in_sizes[2] / 2;
  const int Eq = in_sizes[3] / 2;
  const int* src_d = ei_d;           const int* dst_d = ei_d + Ed;
  const int* src_q = ei_q;           const int* dst_q = ei_q + Eq;
  const float* W1 = (const float*)d_in[6];   const float* b1 = (const float*)d_in[7];
  const float* W2 = (const float*)d_in[8];   const float* b2 = (const float*)d_in[9];
  const float* W3 = (const float*)d_in[10];  const float* b3 = (const float*)d_in[11];
  const float* Wns[3] = {(const float*)d_in[12], (const float*)d_in[17], (const float*)d_in[22]};
  const float* Vns[3] = {(const float*)d_in[13], (const float*)d_in[18], (const float*)d_in[23]};
  const float* bns[3] = {(const float*)d_in[14], (const float*)d_in[19], (const float*)d_in[24]};
  const float* cws[3] = {(const float*)d_in[15], (const float*)d_in[20], (const float*)d_in[25]};
  const float* cbs[3] = {(const float*)d_in[16], (const float*)d_in[21], (const float*)d_in[26]};
  const float* w_end = (const float*)d_in[27];
  const float* b_end = (const float*)d_in[28];

  const int Nd = B_ * ND_;   // 32768
  const int Nq = B_ * NQ_;   // 2048

  char* ws = (char*)d_ws;
  auto alloc = [&](size_t nfloats) {
    float* p = (float*)ws;
    ws += nfloats * sizeof(float);
    return p;
  };
  float* dinv_d = alloc(Nd);
  float* dinv_q = alloc(Nq);
  float* Htmp = alloc((size_t)Nd * D_);
  float* dl[3] = {alloc((size_t)Nd * D_), alloc((size_t)Nd * D_), alloc((size_t)Nd * D_)};
  float* ql[3] = {alloc((size_t)Nq * D_), alloc((size_t)Nq * D_), alloc((size_t)Nq * D_)};
  float* qW  = alloc((size_t)B_ * K_ * NQ_ * D_);
  float* att = alloc((size_t)B_ * NQ_ * ND_);
  float* lqb = alloc((size_t)B_ * K_ * NQ_);
  float* ldb = alloc((size_t)B_ * K_ * ND_);
  float* eraw[3] = {alloc((size_t)B_ * NQ_ * ND_), alloc((size_t)B_ * NQ_ * ND_),
                    alloc((size_t)B_ * NQ_ * ND_)};
  float* dir2 = alloc((size_t)B_ * NQ_ * ND_);
  float* dir3 = alloc((size_t)B_ * NQ_ * ND_);
  float* dirs[3] = {nullptr, dir2, dir3};

  // --- degree normalization (self loop contributes 1) ---
  k_fill<<<(Nd + 255) / 256, 256, 0, stream>>>(dinv_d, 1.0f, Nd);
  k_fill<<<(Nq + 255) / 256, 256, 0, stream>>>(dinv_q, 1.0f, Nq);
  k_deg<<<(Ed + 255) / 256, 256, 0, stream>>>(dst_d, dinv_d, Ed);
  k_deg<<<(Eq + 255) / 256, 256, 0, stream>>>(dst_q, dinv_q, Eq);
  k_dinv<<<(Nd + 255) / 256, 256, 0, stream>>>(dinv_d, Nd);
  k_dinv<<<(Nq + 255) / 256, 256, 0, stream>>>(dinv_q, Nq);

  // --- GCN stack (shared weights, both graphs) ---
  auto gcn_layer = [&](const float* X, int N, int Din, const float* W,
                       const float* bias, const int* src, const int* dst, int E,
                       const float* dinv, float* Out) {
    if (Din == DIN_)
      k_gemm_nn<DIN_><<<dim3(N / 16, D_ / 16), 32, 0, stream>>>(X, W, Htmp);
    else
      k_gemm_nn<D_><<<dim3(N / 16, D_ / 16), 32, 0, stream>>>(X, W, Htmp);
    int tot = N * D_;
    k_gcn_self<<<(tot + 255) / 256, 256, 0, stream>>>(Htmp, dinv, bias, Out, tot);
    int etot = E * D_;
    k_gcn_edge<<<(etot + 255) / 256, 256, 0, stream>>>(src, dst, Htmp, dinv, Out, E);
    k_relu<<<(tot + 255) / 256, 256, 0, stream>>>(Out, tot);
  };
  gcn_layer(x_d, Nd, DIN_, W1, b1, src_d, dst_d, Ed, dinv_d, dl[0]);
  gcn_layer(dl[0], Nd, D_, W2, b2, src_d, dst_d, Ed, dinv_d, dl[1]);
  gcn_layer(dl[1], Nd, D_, W3, b3, src_d, dst_d, Ed, dinv_d, dl[2]);
  gcn_layer(x_q, Nq, DIN_, W1, b1, src_q, dst_q, Eq, dinv_q, ql[0]);
  gcn_layer(ql[0], Nq, D_, W2, b2, src_q, dst_q, Eq, dinv_q, ql[1]);
  gcn_layer(ql[1], Nq, D_, W3, b3, src_q, dst_q, Eq, dinv_q, ql[2]);

  // --- per-level attention + NTN (fused, he never materialized) ---
  for (int l = 0; l < 3; ++l) {
    k_att_logits<<<dim3(NQ_ / 16, ND_ / 16, B_), 32, 0, stream>>>(ql[l], dl[l], att);
    k_softmax_rows512<<<B_ * NQ_, 256, 0, stream>>>(att);
    k_lq<<<(B_ * K_ * NQ_ + 255) / 256, 256, 0, stream>>>(Vns[l], ql[l], lqb);
    k_ld<<<(B_ * K_ * ND_ + 255) / 256, 256, 0, stream>>>(Vns[l], dl[l], ldb);
    k_qw<<<dim3(NQ_ / 16, D_ / 16, B_ * K_), 32, 0, stream>>>(ql[l], Wns[l], qW);
    k_level<<<dim3(NQ_ / 16, ND_ / 16, B_), 512, 0, stream>>>(
        qW, dl[l], att, lqb, ldb, bns[l], cws[l], cbs[l], w_end, l + 1,
        eraw[l], dirs[l]);
  }

  // --- final combine + softmax ---
  k_softmax_rows512<<<B_ * NQ_, 256, 0, stream>>>(eraw[0]);
  k_softmax_rows512<<<B_ * NQ_, 256, 0, stream>>>(eraw[2]);
  int tot = B_ * NQ_ * ND_;
  k_combine<<<(tot + 255) / 256, 256, 0, stream>>>(eraw[0], eraw[1], eraw[2],
                                                   dir2, dir3, w_end, b_end,
                                                   (float*)d_out, tot);
  k_softmax_rows512<<<B_ * NQ_, 256, 0, stream>>>((float*)d_out);
}